// LearnableKernel_16793322127721
// MI455X (gfx1250) — compile-verified
//
#include <hip/hip_runtime.h>
#include <hip/hip_bf16.h>

#define BB 4
#define CH 128
#define PC 64
#define NN 4096

typedef float v2f __attribute__((ext_vector_type(2)));
typedef float v8f __attribute__((ext_vector_type(8)));

// ---------------------------------------------------------------------------
// Kernel A: pos = Wp @ pos_bot + bp, stored transposed as x[B,N,64] (row major
// per position => exactly the per-lane layout WMMA fragments want), plus
// sq[B,N] = ||x_row||^2.
// ---------------------------------------------------------------------------
__global__ __launch_bounds__(128) void pos_proj_kernel(
    const float* __restrict__ pos_bot,  // [B,64,N]
    const float* __restrict__ Wp,       // [64,64]
    const float* __restrict__ bp,       // [64]
    float* __restrict__ x,              // [B,N,64]
    float* __restrict__ sq)             // [B,N]
{
    __shared__ float sW[PC * PC];       // 16 KB
    for (int i = threadIdx.x; i < PC * PC; i += 128) sW[i] = Wp[i];
    __syncthreads();

    const int g = blockIdx.x * 128 + threadIdx.x;   // [0, B*N)
    const int b = g >> 12;
    const int n = g & (NN - 1);

    float in[PC];
    const float* src = pos_bot + (size_t)b * PC * NN + n;
#pragma unroll
    for (int c = 0; c < PC; ++c) in[c] = src[(size_t)c * NN];

    float s = 0.0f;
    float* xrow = x + (size_t)g * PC;
    for (int o = 0; o < PC; ++o) {
        float acc = bp[o];
        const float* w = &sW[o * PC];
#pragma unroll 16
        for (int c = 0; c < PC; ++c) acc = fmaf(w[c], in[c], acc);
        xrow[o] = acc;
        s = fmaf(acc, acc, s);
    }
    sq[g] = s;
}

// ---------------------------------------------------------------------------
// Kernel B: equation_F[b,n,o] = (Wv @ corr_feats + bv) transposed, written
// straight into the output tail.
// ---------------------------------------------------------------------------
__global__ __launch_bounds__(128) void value_proj_kernel(
    const float* __restrict__ corr,     // [B,128,N]
    const float* __restrict__ Wv,       // [128,128]
    const float* __restrict__ bv,       // [128]
    float* __restrict__ eqF)            // [B,N,128]
{
    __shared__ float sW[CH * CH];       // 64 KB (WGP has 320 KB)
    for (int i = threadIdx.x; i < CH * CH; i += 128) sW[i] = Wv[i];
    __syncthreads();

    const int g = blockIdx.x * 128 + threadIdx.x;   // [0, B*N)
    const int b = g >> 12;
    const int n = g & (NN - 1);

    float in[CH];
    const float* src = corr + (size_t)b * CH * NN + n;
#pragma unroll
    for (int c = 0; c < CH; ++c) in[c] = src[(size_t)c * NN];

    float* dst = eqF + (size_t)g * CH;
    for (int o = 0; o < CH; ++o) {
        float acc = bv[o];
        const float* w = &sW[o * CH];
#pragma unroll 16
        for (int c = 0; c < CH; ++c) acc = fmaf(w[c], in[c], acc);
        dst[o] = acc;
    }
}

// ---------------------------------------------------------------------------
// Kernel C: 128x128 output block per 8-wave workgroup.
//   stage:   A panel (128 rows) + B panel (128 rows) of x -> LDS (K=64 whole)
//   compute: wave w owns rows [w*16, w*16+16) x 128 cols: 8 v8f accumulators,
//            128 x V_WMMA_F32_16X16X4_F32, fragments via ds_load_b64
//   epilogue: out = exp(-beta * max(0, sq_i + sq_j - 2*G)), coalesced stores
//
// ISA fragment layouts (32-bit, wave32):
//   A 16x4 : lane L holds M = L%16 ; VGPR v holds K = v + 2*(L/16)
//   B 4x16 : lane L holds N = L%16 ; VGPR v holds K = v + 2*(L/16)
//   C 16x16: VGPR r, lane L -> (row = r + 8*(L/16), col = L%16)
// LDS row stride 68 floats: 16B-aligned rows, and fragment b64 loads hit
// banks (4*row + ofs) mod 64 -> all 64 banks exactly once (conflict-free).
// ---------------------------------------------------------------------------
__global__ __launch_bounds__(256) void gram_rbf_kernel(
    const float* __restrict__ x,        // [B,N,64]
    const float* __restrict__ sq,       // [B,N]
    const float* __restrict__ beta_p,   // [1]
    float* __restrict__ out)            // [B,N,N]
{
    constexpr int STR = 68;             // padded LDS row stride (floats)
    __shared__ float sA[128 * STR];     // 34816 B
    __shared__ float sB[128 * STR];     // 34816 B

    const int tid  = threadIdx.x;
    const int lane = tid & 31;
    const int w    = tid >> 5;          // wave id 0..7
    const int lm   = lane & 15;
    const int lh   = lane >> 4;

    const int bx      = blockIdx.x;     // 4096 blocks: b * 32 * 32
    const int b       = bx >> 10;
    const int rem     = bx & 1023;
    const int rowBase = (rem >> 5) << 7;    // rowBlk * 128
    const int colBase = (rem & 31) << 7;    // colBlk * 128

    const float* xb  = x  + (size_t)b * NN * PC;
    const float* sqb = sq + (size_t)b * NN;

    // ---- cooperative stage: 2 x (128 rows x 64 floats) -> LDS -------------
    for (int i = tid; i < 128 * 16; i += 256) {
        const int row = i >> 4;
        const int c4  = (i & 15) << 2;
        const float4 va = *(const float4*)(xb + (size_t)(rowBase + row) * PC + c4);
        const float4 vb = *(const float4*)(xb + (size_t)(colBase + row) * PC + c4);
        *(float4*)&sA[row * STR + c4] = va;
        *(float4*)&sB[row * STR + c4] = vb;
    }
    __syncthreads();

    // ---- WMMA main loop ---------------------------------------------------
    v8f acc[8];
    const v8f zero = {};
#pragma unroll
    for (int t = 0; t < 8; ++t) acc[t] = zero;

    const float* aRow = &sA[(w * 16 + lm) * STR + 2 * lh];
    const float* bRow = &sB[lm * STR + 2 * lh];

#pragma unroll
    for (int k = 0; k < PC; k += 4) {
        const v2f a = *(const v2f*)(aRow + k);
#pragma unroll
        for (int t = 0; t < 8; ++t) {
            const v2f bf = *(const v2f*)(bRow + t * 16 * STR + k);
            acc[t] = __builtin_amdgcn_wmma_f32_16x16x4_f32(
                /*neg_a=*/false, a, /*neg_b=*/false, bf,
                /*c_mod=*/(short)0, acc[t],
                /*reuse_a=*/false, /*reuse_b=*/false);
        }
    }

    // ---- fused RBF epilogue ----------------------------------------------
    const float beta = beta_p[0];
    float sr[8];
#pragma unroll
    for (int r = 0; r < 8; ++r) sr[r] = sqb[rowBase + w * 16 + r + 8 * lh];

#pragma unroll
    for (int t = 0; t < 8; ++t) {
        const float sqc = sqb[colBase + t * 16 + lm];
        const int   col = colBase + t * 16 + lm;
#pragma unroll
        for (int r = 0; r < 8; ++r) {
            const int row = rowBase + w * 16 + r + 8 * lh;
            float d2 = sr[r] + sqc - 2.0f * acc[t][r];
            d2 = fmaxf(d2, 0.0f);
            out[((size_t)b * NN + row) * NN + col] = __expf(-beta * d2);
        }
    }
}

// ---------------------------------------------------------------------------
extern "C" void kernel_launch(void* const* d_in, const int* in_sizes, int n_in,
                              void* d_out, int out_size, void* d_ws, size_t ws_size,
                              hipStream_t stream) {
    const float* pos_bot = (const float*)d_in[0];   // [4,64,4096]
    const float* corr    = (const float*)d_in[1];   // [4,128,4096]
    const float* Wp      = (const float*)d_in[2];   // [64,64]
    const float* bp      = (const float*)d_in[3];   // [64]
    const float* Wv      = (const float*)d_in[4];   // [128,128]
    const float* bv      = (const float*)d_in[5];   // [128]
    const float* beta    = (const float*)d_in[6];   // [1]

    float* out = (float*)d_out;                     // kernel [4,4096,4096]
    float* eqF = out + (size_t)BB * NN * NN;        // equation_F [4,4096,128]

    float* x  = (float*)d_ws;                       // [4,4096,64]  (4 MB)
    float* sq = x + (size_t)BB * NN * PC;           // [4,4096]     (64 KB)

    pos_proj_kernel<<<(BB * NN) / 128, 128, 0, stream>>>(pos_bot, Wp, bp, x, sq);
    value_proj_kernel<<<(BB * NN) / 128, 128, 0, stream>>>(corr, Wv, bv, eqF);

    const int blocks = BB * (NN / 128) * (NN / 128);    // 4096
    gram_rbf_kernel<<<blocks, 256, 0, stream>>>(x, sq, beta, out);
}